// DeepVoxels_63419487093183
// MI455X (gfx1250) — compile-verified
//
#include <hip/hip_runtime.h>
#include <stdint.h>

#define NFEAT 64
#define OCTL  299592          // octree length per feature
#define EPSF  1e-5f
#define CHUNK 16384           // floats per TDM tile (64 KB LDS)

typedef __attribute__((ext_vector_type(4))) unsigned int u32x4;
typedef __attribute__((ext_vector_type(8))) int i32x8;
typedef __attribute__((ext_vector_type(4))) int i32x4;

// ---------------------------------------------------------------------------
// Bulk copy src -> dst through the Tensor Data Mover.
// Wave 0 of each block DMAs a 64KB tile global->LDS (TENSORcnt-tracked),
// then all 8 waves stream LDS -> global with b128 stores.
// ---------------------------------------------------------------------------
__global__ void copy_tdm_kernel(const float* __restrict__ src,
                                float* __restrict__ dst, int ntotal) {
  __shared__ float tile[CHUNK];
  const int base = blockIdx.x * CHUNK;
  int n = ntotal - base;
  if (n > CHUNK) n = CHUNK;

  if (threadIdx.x < 32) {   // single wave issues the TDM op
    const uint64_t gaddr = (uint64_t)(uintptr_t)src + (uint64_t)base * 4u;
    const unsigned int ldsOff = (unsigned int)(uintptr_t)(void*)tile; // low 32b = LDS byte offset

    u32x4 g0;
    g0.x = 1u;                                          // count = 1 (valid D#)
    g0.y = ldsOff;                                      // lds_addr (bytes)
    g0.z = (unsigned int)(gaddr & 0xFFFFFFFFull);       // global_addr[31:0]
    g0.w = (unsigned int)((gaddr >> 32) & 0x01FFFFFFull)// global_addr[56:32]
         | (2u << 30);                                  // type = 2 ("image")

    i32x8 g1;
    g1.s0 = (int)(2u << 16);                 // workgroup_mask=0, data_size=2 (4 bytes)
    g1.s1 = (int)(((unsigned)n & 0xFFFFu) << 16);        // tensor_dim0[15:0]
    g1.s2 = (int)(((unsigned)n >> 16) & 0xFFFFu) | (1 << 16); // tensor_dim0[31:16] | tensor_dim1=1
    g1.s3 = (int)((unsigned)n << 16);        // tensor_dim1 hi=0 | tile_dim0 = n (<=16384)
    g1.s4 = 1 | (1 << 16);                   // tile_dim1 = 1 | tile_dim2 = 1
    g1.s5 = n;                               // tensor_dim0_stride lo32 = n
    g1.s6 = 0;                               // stride hi | tensor_dim1_stride lo = 0
    g1.s7 = 0;

    i32x4 g2; g2.x = 1; g2.y = 1; g2.z = 0; g2.w = (1 << 16); // tdim2=1, tdim3=1, tile_dim3=1
    i32x4 g3; g3.x = 0; g3.y = (1 << 16); g3.z = (1 << 16); g3.w = 0; // tdim4=1, tile_dim4=1
    i32x8 gz = {0, 0, 0, 0, 0, 0, 0, 0};     // uncharacterized 5th operand (zero-filled)

    __builtin_amdgcn_tensor_load_to_lds(g0, g1, g2, g3, gz, 0);
    __builtin_amdgcn_s_wait_tensorcnt(0);
  }
  __syncthreads();

  const int n4 = n >> 2;                                   // n is always a multiple of 4
  float4* __restrict__ d4 = (float4*)(dst + base);
  const float4* __restrict__ t4 = (const float4*)tile;
  for (int k = threadIdx.x; k < n4; k += blockDim.x)
    d4[k] = t4[k];
}

// ---------------------------------------------------------------------------
// octree2voxel prune level (fully parallel; races are benign same-value-0
// writes, every same-level write preserves the <EPS predicate it may feed).
//   mask_j = (oct[j+dl] < eps) | (j>=start && oct[p+dl] < eps), p=(j-start)>>3
//   if mask: zero node j+dl and its 8 children at end+8j (32B-aligned).
// ---------------------------------------------------------------------------
__global__ void prune_level_kernel(float* __restrict__ oct,
                                   int dl, int start, int end, int shift) {
  const int idx = blockIdx.x * blockDim.x + threadIdx.x;
  const int j = idx & (dl - 1);
  const int f = idx >> shift;
  if (f >= NFEAT) return;
  const size_t fb = (size_t)f * OCTL;

  bool m = oct[fb + dl + j] < EPSF;
  if (j >= start) {
    const int p = (j - start) >> 3;
    m = m | (oct[fb + dl + p] < EPSF);
  }
  if (m) {
    oct[fb + dl + j] = 0.0f;
    const float4 z = make_float4(0.f, 0.f, 0.f, 0.f);
    float4* cp = (float4*)(oct + fb + end + 8 * (size_t)j);
    cp[0] = z;
    cp[1] = z;
  }
}

// ---------------------------------------------------------------------------
// voxel2octree mean level: oct[dl+j] = mean(oct[end+8j .. end+8j+7]).
// Children below index 8*dl are logically zero (concat zeros) for k<4, so
// j < cutoff writes 0 without reading; j >= cutoff reads only the previous
// (finished) level's node region -> no intra-level hazard. Level k=4 uses
// cutoff=0 and is split into two launches at j=4680 to break its hazard.
// ---------------------------------------------------------------------------
__global__ void mean_level_kernel(float* __restrict__ oct,
                                  int jbase, int jend, int dl, int end, int cutoff) {
  const int j = jbase + blockIdx.x * blockDim.x + threadIdx.x;
  if (j >= jend) return;
  const size_t fb = (size_t)blockIdx.y * OCTL;

  float out = 0.0f;
  if (j >= cutoff) {
    const float4* cp = (const float4*)(oct + fb + end + 8 * (size_t)j);
    const float4 a = cp[0];
    const float4 b = cp[1];
    out = ((((((a.x + a.y) + a.z) + a.w) + b.x) + b.y) + b.z) + b.w;
    out *= 0.125f;
  }
  oct[fb + dl + j] = out;
}

// Zero positions [0, 32768) per feature (the "concat zeros" region; the mean
// levels then overwrite their node sub-ranges). 16B-vectorized.
__global__ void zero_head_kernel(float* __restrict__ oct) {
  const int q4 = blockIdx.x * blockDim.x + threadIdx.x;   // [0, 8192)
  const size_t fb = (size_t)blockIdx.y * OCTL;
  const float4 z = make_float4(0.f, 0.f, 0.f, 0.f);
  ((float4*)(oct + fb))[q4] = z;
}

extern "C" void kernel_launch(void* const* d_in, const int* in_sizes, int n_in,
                              void* d_out, int out_size, void* d_ws, size_t ws_size,
                              hipStream_t stream) {
  (void)in_sizes; (void)n_in; (void)out_size; (void)d_ws; (void)ws_size;
  const float* src = (const float*)d_in[0];
  float* oct = (float*)d_out;
  const int N = NFEAT * OCTL;   // 19,173,888 floats

  // 1) copy input -> d_out via TDM (we must not mutate d_in)
  copy_tdm_kernel<<<(N + CHUNK - 1) / CHUNK, 256, 0, stream>>>(src, oct, N);

  // 2) octree2voxel pruning, levels i = 0..4 (stream-ordered)
  const int DLs[5]    = {8,  64,  512, 4096, 32768};
  const int STARTs[5] = {0,   8,   72,  584,  4680};
  const int ENDs[5]   = {8,  72,  584, 4680, 37448};
  const int SHIFTs[5] = {3,   6,    9,   12,    15};
  for (int i = 0; i < 5; ++i) {
    const int threads = NFEAT * DLs[i];
    prune_level_kernel<<<(threads + 255) / 256, 256, 0, stream>>>(
        oct, DLs[i], STARTs[i], ENDs[i], SHIFTs[i]);
  }

  // 3) voxel2octree means, deepest (k=4) split at j=4680 for hazard safety
  mean_level_kernel<<<dim3( 19, NFEAT), 256, 0, stream>>>(oct,    0,  4680, 32768, 37448,    0);
  mean_level_kernel<<<dim3(110, NFEAT), 256, 0, stream>>>(oct, 4680, 32768, 32768, 37448,    0);
  zero_head_kernel <<<dim3( 32, NFEAT), 256, 0, stream>>>(oct);
  mean_level_kernel<<<dim3( 16, NFEAT), 256, 0, stream>>>(oct,    0,  4096,  4096,  4680, 3511);
  mean_level_kernel<<<dim3(  2, NFEAT), 256, 0, stream>>>(oct,    0,   512,   512,   584,  439);
  mean_level_kernel<<<dim3(  1, NFEAT), 256, 0, stream>>>(oct,    0,    64,    64,    72,   55);
  mean_level_kernel<<<dim3(  1, NFEAT), 256, 0, stream>>>(oct,    0,     8,     8,     8,    7);
}